// SAGE_57543971832597
// MI455X (gfx1250) — compile-verified
//
#include <hip/hip_runtime.h>
#include <math.h>

typedef __attribute__((ext_vector_type(2))) float v2f;
typedef __attribute__((ext_vector_type(8))) float v8f;

#define N3c 512
#define N2c 3072
#define N1c 33792
#define N0c 540672
#define INc 128
#define HIDc 256
#define OUTc 64

#if defined(__HIP_DEVICE_COMPILE__) && \
    __has_builtin(__builtin_amdgcn_global_load_async_to_lds_b128) && \
    __has_builtin(__builtin_amdgcn_s_wait_asynccnt)
#define HAVE_ASYNC_LDS 1
// Builtin signature (from hipcc diagnostic): non-const int4* (vector_size(16))
// in the global (__device__/AS1) address space, plus an AS3 LDS pointer.
typedef int i4vec __attribute__((vector_size(16)));
typedef __attribute__((address_space(1))) i4vec g_i4;
typedef __attribute__((address_space(3))) i4vec l_i4;
#else
#define HAVE_ASYNC_LDS 0
#endif

// ---------------------------------------------------------------------------
// Gather + mean aggregation. Edges are pre-grouped: dst d owns src[d*FAN .. d*FAN+FAN).
// One wave per destination node; lanes hold the feature vector as float4s.
// Indices are preloaded one-per-lane and broadcast via shfl so the fully
// unrolled fanout loop can put all FAN row loads in flight simultaneously.
// ---------------------------------------------------------------------------
template<int FIN, int FAN>
__global__ __launch_bounds__(256) void sage_agg_kernel(
    const float* __restrict__ x_src,   // [n_src, FIN]
    const int*   __restrict__ src_idx, // [n_dst*FAN]
    float*       __restrict__ agg,     // [n_dst, FIN]
    int n_dst)
{
    constexpr int V4 = FIN / 128;      // float4s per lane (1 or 2)
    int wave = (int)((blockIdx.x * blockDim.x + threadIdx.x) >> 5);
    int lane = (int)(threadIdx.x & 31);
    if (wave >= n_dst) return;

    const int* srcs = src_idx + (size_t)wave * FAN;
    int myidx = (lane < FAN) ? srcs[lane] : 0;   // one neighbor index per lane

    float4 acc[V4];
#pragma unroll
    for (int v = 0; v < V4; ++v) { acc[v].x = 0.f; acc[v].y = 0.f; acc[v].z = 0.f; acc[v].w = 0.f; }

#pragma unroll
    for (int j = 0; j < FAN; ++j) {
        int s = __shfl(myidx, j, 32);  // wave-uniform neighbor index
        const float4* row4 = (const float4*)(x_src + (size_t)s * FIN);
#pragma unroll
        for (int v = 0; v < V4; ++v) {
            float4 t = row4[lane + 32 * v];
            acc[v].x += t.x; acc[v].y += t.y; acc[v].z += t.z; acc[v].w += t.w;
        }
    }
    const float cnt = (float)FAN;      // exact match of reference agg/cnt
    float4* o4 = (float4*)(agg + (size_t)wave * FIN);
#pragma unroll
    for (int v = 0; v < V4; ++v) {
        float4 t;
        t.x = acc[v].x / cnt; t.y = acc[v].y / cnt; t.z = acc[v].z / cnt; t.w = acc[v].w / cnt;
        o4[lane + 32 * v] = t;
    }
}

// ---------------------------------------------------------------------------
// out[row0:row0+16, :] = [agg | x_dst] @ [Wl ; Wr] + b   (+ optional ReLU)
// Block = (FOUT/16) waves; wave w owns output columns [16w,16w+16).
// A panel (16 x 2*FIN) staged in LDS (async-to-LDS when available);
// K-loop of V_WMMA_F32_16X16X4_F32, split into Wl/Wr halves.
// ---------------------------------------------------------------------------
template<int FIN, int FOUT>
__global__ __launch_bounds__((FOUT / 16) * 32) void sage_gemm_kernel(
    const float* __restrict__ agg,   // [n_dst, FIN]
    const float* __restrict__ xdst,  // rows 0..n_dst-1 of source features, stride FIN
    const float* __restrict__ Wl,    // [FIN, FOUT]
    const float* __restrict__ Wr,    // [FIN, FOUT]
    const float* __restrict__ bias,  // [FOUT]
    float*       __restrict__ out,   // [n_dst, FOUT]
    int relu)
{
    constexpr int NW   = FOUT / 16;
    constexpr int KTOT = 2 * FIN;
    __shared__ __align__(16) float Atile[16 * KTOT];   // [16][KTOT]: agg then xdst

    const int lane = (int)(threadIdx.x & 31);
    const int wave = (int)(threadIdx.x >> 5);
    const int row0 = (int)blockIdx.x * 16;
    const int tid  = (int)threadIdx.x;
    constexpr int NTHREADS = NW * 32;

    // Cooperative A-panel staging (float4 granularity).
    for (int idx = tid; idx < 16 * (FIN / 4); idx += NTHREADS) {
        int r  = idx / (FIN / 4);
        int c4 = idx % (FIN / 4);
        const float4* gva = ((const float4*)(agg  + (size_t)(row0 + r) * FIN)) + c4;
        const float4* gvx = ((const float4*)(xdst + (size_t)(row0 + r) * FIN)) + c4;
        float4* la = (float4*)&Atile[r * KTOT + c4 * 4];
        float4* lx = (float4*)&Atile[r * KTOT + FIN + c4 * 4];
#if HAVE_ASYNC_LDS
        __builtin_amdgcn_global_load_async_to_lds_b128((g_i4*)gva, (l_i4*)la, 0, 0);
        __builtin_amdgcn_global_load_async_to_lds_b128((g_i4*)gvx, (l_i4*)lx, 0, 0);
#else
        *la = *gva;
        *lx = *gvx;
#endif
    }
#if HAVE_ASYNC_LDS
    __builtin_amdgcn_s_wait_asynccnt(0);
#endif
    __syncthreads();

    const int col0 = wave * 16;
    const int arow = lane & 15;          // A: M = lane (lanes 0-15); lanes 16-31 same M, K+2
    const int kof  = (lane >> 4) * 2;
    const int colc = lane & 15;          // B/C/D: N striped across lanes within a VGPR

    v8f c = {};
    const float* Whalf[2] = { Wl, Wr };
#pragma unroll
    for (int half = 0; half < 2; ++half) {
        const float* __restrict__ W = Whalf[half];
        const int kbase = half * FIN;
#pragma unroll 4
        for (int k = 0; k < FIN; k += 4) {
            // A fragment: 2 consecutive K values for this lane's row (8B-aligned LDS read).
            v2f a = *(const v2f*)&Atile[arow * KTOT + kbase + k + kof];
            int kk = k + kof;
            v2f b;
            b.x = W[(size_t)kk * FOUT + col0 + colc];
            b.y = W[(size_t)kk * FOUT + FOUT + col0 + colc];
            c = __builtin_amdgcn_wmma_f32_16x16x4_f32(
                    /*neg_a=*/false, a, /*neg_b=*/false, b,
                    /*c_mod=*/(short)0, c, /*reuse_a=*/false, /*reuse_b=*/false);
        }
    }

    // Store: VGPR r holds M=r (lanes 0-15) or M=r+8 (lanes 16-31), N = lane&15.
    const int ocol  = col0 + colc;
    const float bv  = bias[ocol];
    const int mbase = (lane >> 4) * 8;
#pragma unroll
    for (int r = 0; r < 8; ++r) {
        float v = c[r] + bv;
        if (relu) v = fmaxf(v, 0.f);
        out[(size_t)(row0 + mbase + r) * FOUT + ocol] = v;
    }
}

// ---------------------------------------------------------------------------
// Row-wise log_softmax over 64 columns: one wave per row, 2 values per lane.
// ---------------------------------------------------------------------------
__global__ __launch_bounds__(256) void logsoftmax64_kernel(
    const float* __restrict__ h, float* __restrict__ out, int nrows)
{
    int wave = (int)((blockIdx.x * blockDim.x + threadIdx.x) >> 5);
    int lane = (int)(threadIdx.x & 31);
    if (wave >= nrows) return;

    float2 v = ((const float2*)(h + (size_t)wave * 64))[lane];
    float m = fmaxf(v.x, v.y);
#pragma unroll
    for (int off = 16; off > 0; off >>= 1) m = fmaxf(m, __shfl_xor(m, off, 32));
    float e = expf(v.x - m) + expf(v.y - m);
#pragma unroll
    for (int off = 16; off > 0; off >>= 1) e += __shfl_xor(e, off, 32);
    float lse = logf(e) + m;
    float2 o; o.x = v.x - lse; o.y = v.y - lse;
    ((float2*)(out + (size_t)wave * 64))[lane] = o;
}

// ---------------------------------------------------------------------------
extern "C" void kernel_launch(void* const* d_in, const int* in_sizes, int n_in,
                              void* d_out, int out_size, void* d_ws, size_t ws_size,
                              hipStream_t stream)
{
    const float* x   = (const float*)d_in[0];
    const int*   ei0 = (const int*)  d_in[1];   // [2, N1*15] : src row first
    const int*   ei1 = (const int*)  d_in[2];   // [2, N2*10]
    const int*   ei2 = (const int*)  d_in[3];   // [2, N3*5]
    const float* Wl0 = (const float*)d_in[4];
    const float* bl0 = (const float*)d_in[5];
    const float* Wr0 = (const float*)d_in[6];
    const float* Wl1 = (const float*)d_in[7];
    const float* bl1 = (const float*)d_in[8];
    const float* Wr1 = (const float*)d_in[9];
    const float* Wl2 = (const float*)d_in[10];
    const float* bl2 = (const float*)d_in[11];
    const float* Wr2 = (const float*)d_in[12];
    float* outp = (float*)d_out;

    float* ws = (float*)d_ws;
    size_t off = 0;
    float* agg0 = ws + off; off += (size_t)N1c * INc;   // 33792*128
    float* h1   = ws + off; off += (size_t)N1c * HIDc;  // 33792*256
    float* agg1 = ws + off; off += (size_t)N2c * HIDc;  // 3072*256
    float* h2   = ws + off; off += (size_t)N2c * HIDc;  // 3072*256
    float* agg2 = ws + off; off += (size_t)N3c * HIDc;  // 512*256
    float* h3   = ws + off; off += (size_t)N3c * OUTc;  // 512*64
    (void)ws_size; (void)in_sizes; (void)n_in; (void)out_size;

    // ----- Layer 0: N0(128) -> N1(256), ReLU -----
    sage_agg_kernel<INc, 15><<<N1c / 8, 256, 0, stream>>>(x, ei0, agg0, N1c);
    sage_gemm_kernel<INc, HIDc><<<N1c / 16, (HIDc / 16) * 32, 0, stream>>>(
        agg0, x, Wl0, Wr0, bl0, h1, /*relu=*/1);

    // ----- Layer 1: N1(256) -> N2(256), ReLU -----
    sage_agg_kernel<HIDc, 10><<<N2c / 8, 256, 0, stream>>>(h1, ei1, agg1, N2c);
    sage_gemm_kernel<HIDc, HIDc><<<N2c / 16, (HIDc / 16) * 32, 0, stream>>>(
        agg1, h1, Wl1, Wr1, bl1, h2, /*relu=*/1);

    // ----- Layer 2: N2(256) -> N3(64), log_softmax -----
    sage_agg_kernel<HIDc, 5><<<N3c / 8, 256, 0, stream>>>(h2, ei2, agg2, N3c);
    sage_gemm_kernel<HIDc, OUTc><<<N3c / 16, (OUTc / 16) * 32, 0, stream>>>(
        agg2, h2, Wl2, Wr2, bl2, h3, /*relu=*/0);
    logsoftmax64_kernel<<<(N3c + 7) / 8, 256, 0, stream>>>(h3, outp, N3c);
}